// Jaccard_30966714204224
// MI455X (gfx1250) — compile-verified
//
#include <hip/hip_runtime.h>
#include <hip/hip_bf16.h>

#define N_NODES 100000
#define N_EDGES 800000
#define D_IN    128
#define D_HID   256
#define D_OUT   64

typedef __attribute__((ext_vector_type(2))) float v2f;
typedef __attribute__((ext_vector_type(8))) float v8f;

// ---------------------------------------------------------------------------
// Degree / normalization
// ---------------------------------------------------------------------------
__global__ void k_init_deg(float* __restrict__ deg, int n) {
    int i = blockIdx.x * blockDim.x + threadIdx.x;
    if (i < n) deg[i] = 1.0f;  // self-loop contributes 1 to every node's degree
}

__global__ void k_accum_deg(const long long* __restrict__ edst,
                            float* __restrict__ deg, int e) {
    int i = blockIdx.x * blockDim.x + threadIdx.x;
    if (i < e) {
        int d = (int)edst[i];
        __hip_atomic_fetch_add(&deg[d], 1.0f, __ATOMIC_RELAXED,
                               __HIP_MEMORY_SCOPE_AGENT);
    }
}

__global__ void k_to_dinv(float* __restrict__ deg, int n) {
    int i = blockIdx.x * blockDim.x + threadIdx.x;
    if (i < n) deg[i] = rsqrtf(deg[i]);  // deg >= 1 always (self-loop)
}

// ---------------------------------------------------------------------------
// Self-loop init (x4 vectorized): out[i,f] = dinv[i]^2 * src[i,f] (+ bias[f])
// ---------------------------------------------------------------------------
template <int F>
__global__ void k_selfloop_init(const float* __restrict__ src,
                                const float* __restrict__ dinv,
                                const float* __restrict__ bias,
                                float* __restrict__ out, int total4) {
    int i = blockIdx.x * blockDim.x + threadIdx.x;
    if (i >= total4) return;
    const int base = i * 4;
    const int node = base / F;   // F is a power of two -> shift
    const int f    = base % F;
    const float dv = dinv[node];
    const float sc = dv * dv;
#pragma unroll
    for (int j = 0; j < 4; ++j) {
        float v = sc * src[base + j];
        if (bias) v += bias[f + j];
        out[base + j] = v;
    }
}

// ---------------------------------------------------------------------------
// Edge scatter: out[dst] += dinv[src]*dinv[dst] * feat[src]
// 8 waves / block, one wave per edge; each lane owns a contiguous F/32 chunk
// so the gather vectorizes to b128/b64 loads; atomics stay wave-coalesced.
// ---------------------------------------------------------------------------
template <int F>
__global__ __launch_bounds__(256) void k_scatter(const float* __restrict__ feat,
                                                 const long long* __restrict__ esrc,
                                                 const long long* __restrict__ edst,
                                                 const float* __restrict__ dinv,
                                                 float* __restrict__ out) {
    constexpr int EPL = F / 32;  // elements per lane (4 for F=128, 2 for F=64)
    const int wave = threadIdx.x >> 5;
    const int lane = threadIdx.x & 31;
    const long long e = (long long)blockIdx.x * 8 + wave;
    if (e >= N_EDGES) return;

    const int s = (int)esrc[e];
    const int d = (int)edst[e];
    const float nrm = dinv[s] * dinv[d];
    const float* srow = feat + (size_t)s * F + lane * EPL;
    float*       drow = out  + (size_t)d * F + lane * EPL;

    float v[EPL];
#pragma unroll
    for (int j = 0; j < EPL; ++j) v[j] = srow[j];   // contiguous -> b128/b64
#pragma unroll
    for (int j = 0; j < EPL; ++j) {
        __hip_atomic_fetch_add(&drow[j], nrm * v[j], __ATOMIC_RELAXED,
                               __HIP_MEMORY_SCOPE_AGENT);
    }
}

// ---------------------------------------------------------------------------
// WMMA fp32 GEMM: D[M,NC] = A[M,K] @ B[K,NC] (+bias, +relu)
// 8 waves / block; each wave computes a 16(M) x 16*NT(N) strip, reusing one
// A fragment across NT back-to-back V_WMMA_F32_16X16X4_F32 ops per K-step.
// A layout (16x4 fp32): lanes 0-15 hold {K=k,k+1} of row M=lane;
//                       lanes 16-31 hold {K=k+2,k+3} of row M=lane-16.
// B layout (4x16):      mirrored; C/D: VGPR v <-> M = v + 8*(lane>=16).
// ---------------------------------------------------------------------------
template <int K, int NC, int NT, bool RELU>
__global__ __launch_bounds__(256) void k_gemm_wmma(const float* __restrict__ A,
                                                   const float* __restrict__ B,
                                                   const float* __restrict__ bias,
                                                   float* __restrict__ D) {
    const int wave = threadIdx.x >> 5;
    const int lane = threadIdx.x & 31;
    const int m0   = (blockIdx.x * 8 + wave) * 16;
    if (m0 >= N_NODES) return;                    // wave-uniform: EXEC stays full
    const int n0   = blockIdx.y * (16 * NT);
    const int half = lane >> 4;                   // 0: K pair {k,k+1}, 1: {k+2,k+3}
    const int l    = lane & 15;

    const float* arow = A + (size_t)(m0 + l) * K;

    v8f c[NT];
#pragma unroll
    for (int nt = 0; nt < NT; ++nt) c[nt] = v8f{};

#pragma unroll 2
    for (int k = 0; k < K; k += 4) {
        const int ka = k + half * 2;
        v2f a;
        a.x = arow[ka];
        a.y = arow[ka + 1];
#pragma unroll
        for (int nt = 0; nt < NT; ++nt) {
            const float* bc = B + (size_t)(n0 + nt * 16 + l);
            v2f b;
            b.x = bc[(size_t)ka * NC];
            b.y = bc[(size_t)(ka + 1) * NC];
            c[nt] = __builtin_amdgcn_wmma_f32_16x16x4_f32(
                        /*neg_a=*/false, a, /*neg_b=*/false, b,
                        /*c_mod=*/(short)0, c[nt],
                        /*reuse_a=*/false, /*reuse_b=*/false);
        }
    }

#pragma unroll
    for (int nt = 0; nt < NT; ++nt) {
        const int col = n0 + nt * 16 + l;
        const float bb = bias ? bias[col] : 0.0f;
#pragma unroll
        for (int v = 0; v < 8; ++v) {
            const int row = m0 + v + half * 8;
            float val = c[nt][v] + bb;
            if (RELU) val = fmaxf(val, 0.0f);
            D[(size_t)row * NC + col] = val;
        }
    }
}

// ---------------------------------------------------------------------------
// Launch
// ---------------------------------------------------------------------------
extern "C" void kernel_launch(void* const* d_in, const int* in_sizes, int n_in,
                              void* d_out, int out_size, void* d_ws, size_t ws_size,
                              hipStream_t stream) {
    (void)in_sizes; (void)n_in; (void)out_size; (void)ws_size;

    const float*     x  = (const float*)d_in[0];
    const long long* ei = (const long long*)d_in[1];  // [2, E] int64
    const float*     W1 = (const float*)d_in[2];
    const float*     b1 = (const float*)d_in[3];
    const float*     W2 = (const float*)d_in[4];
    const float*     b2 = (const float*)d_in[5];
    float*           out = (float*)d_out;

    const long long* esrc = ei;
    const long long* edst = ei + N_EDGES;

    // Workspace layout (fp32): dinv[N] | xa[N*128] | h[N*256]; t aliases xa.
    float* dinv = (float*)d_ws;
    float* xa   = dinv + N_NODES;
    float* h    = xa + (size_t)N_NODES * D_IN;
    float* t    = xa;  // xa dead after GEMM1; reuse for layer-2 pre-agg buffer

    const int TB = 256;
    const int MT = N_NODES / 16;               // 6250 M-tiles (exact)
    const int GX = (MT + 7) / 8;               // 8 waves (M-tiles) per block

    // 1) degrees (with self-loops) -> dinv = rsqrt(deg)
    k_init_deg<<<(N_NODES + TB - 1) / TB, TB, 0, stream>>>(dinv, N_NODES);
    k_accum_deg<<<(N_EDGES + TB - 1) / TB, TB, 0, stream>>>(edst, dinv, N_EDGES);
    k_to_dinv<<<(N_NODES + TB - 1) / TB, TB, 0, stream>>>(dinv, N_NODES);

    // 2) layer-1 aggregation on the 128-dim INPUT (A_norm @ x), then GEMM.
    {
        const int tot4 = N_NODES * D_IN / 4;
        k_selfloop_init<D_IN><<<(tot4 + TB - 1) / TB, TB, 0, stream>>>(
            x, dinv, nullptr, xa, tot4);
    }
    k_scatter<D_IN><<<(N_EDGES + 7) / 8, TB, 0, stream>>>(x, esrc, edst, dinv, xa);

    // 3) h = relu((A_norm @ x) @ W1 + b1)     [100000 x 256]
    k_gemm_wmma<D_IN, D_HID, 4, true><<<dim3(GX, D_HID / 64), TB, 0, stream>>>(
        xa, W1, b1, h);

    // 4) t = h @ W2                            [100000 x 64]
    k_gemm_wmma<D_HID, D_OUT, 4, false><<<dim3(GX, 1), TB, 0, stream>>>(
        h, W2, nullptr, t);

    // 5) out = A_norm @ t + b2 (self-loop + bias init, then edge scatter)
    {
        const int tot4 = N_NODES * D_OUT / 4;
        k_selfloop_init<D_OUT><<<(tot4 + TB - 1) / TB, TB, 0, stream>>>(
            t, dinv, b2, out, tot4);
    }
    k_scatter<D_OUT><<<(N_EDGES + 7) / 8, TB, 0, stream>>>(t, esrc, edst, dinv, out);
}